// SIMFLUXModel_vector_70171175682106
// MI455X (gfx1250) — compile-verified
//
#include <hip/hip_runtime.h>
#include <hip/hip_bf16.h>
#include <cstdint>

// SIMFLUX vectorized model evaluation for MI455X (gfx1250).
// Bandwidth-bound streaming kernel (~710 MB total traffic -> ~30 us roofline
// at 23.3 TB/s). One block per emitter b, 256 threads = 256 pixels, loop over
// F patterns with psf data held in registers (6x read reuse). Async-LDS DMA
// (ASYNCcnt) stages the stride-5 psf_deriv input tile; a double-buffered LDS
// tile turns the stride-5 deriv output into coalesced b128 non-temporal
// stores (the 600+ MB output stream must bypass L2).

#define TWO_PI_F    6.28318530717958647692f
#define INV_TWO_PI  0.15915494309189533577f
#define INV_PIXEL   (1.0f / 65.0f)

typedef __attribute__((ext_vector_type(4))) float float4v;
typedef __attribute__((__vector_size__(4 * sizeof(int)))) int v4i;

#if defined(__HIP_DEVICE_COMPILE__) && defined(__gfx1250__) && \
    __has_builtin(__builtin_amdgcn_global_load_async_to_lds_b128) && \
    __has_builtin(__builtin_amdgcn_s_wait_asynccnt)
#define USE_ASYNC_LDS 1
#else
#define USE_ASYNC_LDS 0
#endif

__global__ __launch_bounds__(256) void simflux_model_kernel(
    const float* __restrict__ params,     // (B,5)  x,y,z,I,bg
    const float* __restrict__ mod,        // (B,F,6) kx,ky,kz,depth,phase,relint
    const float* __restrict__ roipos,     // (B,2)
    const float* __restrict__ psf_ev,     // (B,256)
    const float* __restrict__ psf_deriv,  // (B,256,5)
    float* __restrict__ mu_out,           // (B,F,256)
    float* __restrict__ deriv_out,        // (B,F,256,5)
    int F)
{
    const int b = blockIdx.x;
    const int t = threadIdx.x;            // pixel index 0..255

    __shared__ float s_int[8];            // per-pattern intensity
    __shared__ float s_dk[8][3];          // per-pattern depth*cos*relint * k
    __shared__ __align__(16) float s_pd[1280];      // input stage: psf_deriv tile
    __shared__ __align__(16) float s_st[2][1280];   // output stage, ping-pong

    const float* gpd = psf_deriv + (size_t)b * 1280;

    // ---- kick off async DMA of the psf_deriv tile first (hides under trig)
#if USE_ASYNC_LDS
    {
        v4i* g = (v4i*)gpd;          // 16B granules
        v4i* l = (v4i*)(&s_pd[0]);
        // 256 lanes x 16B = 4096 B
        __builtin_amdgcn_global_load_async_to_lds_b128(g + t, l + t, 0, 0);
        // remaining 1024 B (64 granules)
        if (t < 64) {
            __builtin_amdgcn_global_load_async_to_lds_b128(g + 256 + t,
                                                           l + 256 + t, 0, 0);
        }
    }
#endif

    // ---- per-pattern modulation (threads 0..F-1) -------------------------
    if (t < F) {
        const float* m = mod + ((size_t)b * F + t) * 6;
        const float k0 = m[0], k1 = m[1], k2 = m[2];
        const float depth = m[3], phase = m[4], relint = m[5];
        const float x = params[b * 5 + 0] + roipos[b * 2 + 0];
        const float y = params[b * 5 + 1] + roipos[b * 2 + 1];
        const float z = params[b * 5 + 2];
        float v = x * k0 + y * k1 + z * k2;
        v = v - floorf(v * INV_TWO_PI) * TWO_PI_F;    // jnp.mod(., 2*pi)
        const float em = v - phase;
        const float sn = sinf(em);
        const float cs = cosf(em);
        s_int[t] = (1.0f + depth * sn) * relint;
        const float cf = depth * cs * relint;
        s_dk[t][0] = cf * k0;
        s_dk[t][1] = cf * k1;
        s_dk[t][2] = cf * k2;
    }

#if USE_ASYNC_LDS
    __builtin_amdgcn_s_wait_asynccnt(0);
#endif
    __syncthreads();   // publishes s_int/s_dk and (async path) s_pd

    // ---- per-pixel invariants (registers, reused for all F patterns) -----
    const float ev   = __builtin_nontemporal_load(psf_ev + (size_t)b * 256 + t);
    const float phot = params[b * 5 + 3];
    const float bgv  = params[b * 5 + 4];

    float pd0, pd1, pd2;
#if USE_ASYNC_LDS
    pd0 = s_pd[t * 5 + 0] * INV_PIXEL;
    pd1 = s_pd[t * 5 + 1] * INV_PIXEL;
    pd2 = s_pd[t * 5 + 2] * 0.001f;   // psf_deriv channels 3,4 are dead code
#else
    pd0 = __builtin_nontemporal_load(gpd + t * 5 + 0) * INV_PIXEL;
    pd1 = __builtin_nontemporal_load(gpd + t * 5 + 1) * INV_PIXEL;
    pd2 = __builtin_nontemporal_load(gpd + t * 5 + 2) * 0.001f;
#endif

    float* muB = mu_out    + ((size_t)b * F) * 256;
    float* dB  = deriv_out + ((size_t)b * F) * 1280;

    // ---- main loop over patterns -----------------------------------------
    for (int f = 0; f < F; ++f) {
        const float it = s_int[f];
        const float d0 = s_dk[f][0];
        const float d1 = s_dk[f][1];
        const float d2 = s_dk[f][2];

        const float mu = fmaf(it * phot, ev, bgv);
        __builtin_nontemporal_store(mu, muB + (size_t)f * 256 + t);

        const float dI = ev * it;
        const float dx = phot * fmaf(d0, ev, it * pd0);
        const float dy = phot * fmaf(d1, ev, it * pd1);
        const float dz = phot * fmaf(d2, ev, it * pd2);

        // Stage deriv tile in the ping-pong LDS buffer (stride-5 writes are
        // bank-conflict-free: gcd(5,64)=1). One barrier per pattern: the
        // barrier of iteration f+1 guarantees all iteration-f reads of the
        // other buffer completed before it is overwritten at f+2.
        float* st = s_st[f & 1];
        st[t * 5 + 0] = dx;
        st[t * 5 + 1] = dy;
        st[t * 5 + 2] = dz;
        st[t * 5 + 3] = dI;
        st[t * 5 + 4] = 1.0f;
        __syncthreads();

        float* dgf = dB + (size_t)f * 1280;
        const float4v v4 = *(const float4v*)&st[t * 4];            // 4096 B
        __builtin_nontemporal_store(v4, (float4v*)dgf + t);
        const float tail = st[1024 + t];                           // 1024 B
        __builtin_nontemporal_store(tail, dgf + 1024 + t);
    }
}

extern "C" void kernel_launch(void* const* d_in, const int* in_sizes, int n_in,
                              void* d_out, int out_size, void* d_ws, size_t ws_size,
                              hipStream_t stream) {
    const float* params    = (const float*)d_in[0];   // (B,5)
    const float* mod       = (const float*)d_in[1];   // (B,F,6)
    const float* roipos    = (const float*)d_in[2];   // (B,2)
    const float* psf_ev    = (const float*)d_in[3];   // (B,16,16)
    const float* psf_deriv = (const float*)d_in[4];   // (B,16,16,5)

    const int B  = in_sizes[0] / 5;
    const int F  = in_sizes[1] / (B * 6);
    const int MM = in_sizes[3] / B;                   // 256 (= 16*16)
    (void)MM; (void)n_in; (void)out_size; (void)d_ws; (void)ws_size;

    float* mu    = (float*)d_out;                              // (B,F,256)
    float* deriv = mu + (size_t)B * F * 256;                   // (B,F,256,5)

    simflux_model_kernel<<<dim3(B), dim3(256), 0, stream>>>(
        params, mod, roipos, psf_ev, psf_deriv, mu, deriv, F);
}